// MiMoV2FlashTopKRouter_17437567222554
// MI455X (gfx1250) — compile-verified
//
#include <hip/hip_runtime.h>
#include <math.h>

typedef __attribute__((ext_vector_type(2))) float v2f;
typedef __attribute__((ext_vector_type(8))) float v8f;

#define HIDDEN  4096
#define NEXP    256
#define TOPK    8
#define KC      128            // K chunk staged through LDS
#define LSTRIDE 132            // 128 + 4 pad -> conflict-free ds_load_b64
#define BM      32             // token rows per block (two 16-row M tiles)
#define NCHUNK  (HIDDEN / KC)

typedef __attribute__((address_space(3))) void lds_void_t;

// Async global -> LDS copy (GLOBAL_LOAD_ASYNC_TO_LDS_B128, ASYNCcnt-tracked).
__device__ __forceinline__ void async_copy_b128(const float* g, const float* lds_generic) {
    unsigned laddr = (unsigned)(unsigned long long)(lds_void_t*)lds_generic;
    asm volatile("global_load_async_to_lds_b128 %0, %1, off"
                 :: "v"(laddr), "v"(g) : "memory");
}
__device__ __forceinline__ void wait_async0() {
    asm volatile("s_wait_asynccnt 0x0" ::: "memory");
}

// ---------------------------------------------------------------------------
// Kernel 1: logits = H @ W^T in fp32 via V_WMMA_F32_16X16X4_F32.
// Block: 256 threads = 8 waves. Block owns 32 token rows; wave w owns experts
// [32w, 32w+32) as two 16-wide N tiles -> 4 independent accumulator chains
// (2M x 2N) per wave. A staged through double-buffered LDS with async copies.
// ---------------------------------------------------------------------------
__global__ __launch_bounds__(256) void gate_gemm_f32(
    const float* __restrict__ H, const float* __restrict__ W,
    float* __restrict__ logits, int T)
{
    __shared__ float buf[2][BM * LSTRIDE];

    const int tid  = threadIdx.x;
    const int lane = tid & 31;
    const int wave = tid >> 5;                 // 0..7
    const int m0   = blockIdx.x * BM;
    const int n0   = wave * 32;

    const int ml = lane & 15;                  // tile row (M for A, N for B)
    const int kh = (lane >> 4) * 2;            // lanes 0-15: K0,K1; 16-31: K2,K3

    v8f acc00 = {}, acc01 = {}, acc10 = {}, acc11 = {};

    const float* w0 = W + (size_t)(n0 + ml) * HIDDEN + kh;        // B tile 0
    const float* w1 = W + (size_t)(n0 + 16 + ml) * HIDDEN + kh;   // B tile 1

    // This thread's 4 async-copy slots: 32 rows x 32 float4 = 1024 b128 / 256 thr
    int rr[4], cc[4];
    #pragma unroll
    for (int i = 0; i < 4; ++i) {
        const int idx = tid + 256 * i;
        rr[i] = idx >> 5;                       // 0..31
        cc[i] = (idx & 31) * 4;                 // 0..124
    }

    // Preload chunk 0 -> buf[0]
    #pragma unroll
    for (int i = 0; i < 4; ++i)
        async_copy_b128(H + (size_t)(m0 + rr[i]) * HIDDEN + cc[i],
                        &buf[0][rr[i] * LSTRIDE + cc[i]]);

    for (int c = 0; c < NCHUNK; ++c) {
        const int cur = c & 1;
        wait_async0();                          // my async writes landed
        __syncthreads();                        // everyone's landed; prev buf free

        if (c + 1 < NCHUNK) {                   // overlap next chunk's HBM stream
            const int k0n = (c + 1) * KC;
            #pragma unroll
            for (int i = 0; i < 4; ++i)
                async_copy_b128(H + (size_t)(m0 + rr[i]) * HIDDEN + k0n + cc[i],
                                &buf[cur ^ 1][rr[i] * LSTRIDE + cc[i]]);
        }

        const float* a0p = &buf[cur][ml * LSTRIDE + kh];          // M tile 0
        const float* a1p = &buf[cur][(16 + ml) * LSTRIDE + kh];   // M tile 1
        const int kbase = c * KC;

        #pragma unroll 4
        for (int kk = 0; kk < KC; kk += 4) {
            v2f a0 = *(const v2f*)(a0p + kk);
            v2f a1 = *(const v2f*)(a1p + kk);
            v2f b0 = *(const v2f*)(w0 + kbase + kk);
            v2f b1 = *(const v2f*)(w1 + kbase + kk);
            // 4 independent accumulation chains -> XDL pipelining
            acc00 = __builtin_amdgcn_wmma_f32_16x16x4_f32(false, a0, false, b0, (short)0, acc00, false, false);
            acc01 = __builtin_amdgcn_wmma_f32_16x16x4_f32(false, a0, false, b1, (short)0, acc01, false, false);
            acc10 = __builtin_amdgcn_wmma_f32_16x16x4_f32(false, a1, false, b0, (short)0, acc10, false, false);
            acc11 = __builtin_amdgcn_wmma_f32_16x16x4_f32(false, a1, false, b1, (short)0, acc11, false, false);
        }
        // no trailing barrier needed: next iteration's top barrier protects buf reuse
    }

    // C/D layout: VGPR r -> row = r + 8*(lane>=16), col = lane&15
    const int mr = 8 * (lane >> 4);
    #pragma unroll
    for (int r = 0; r < 8; ++r) {
        logits[(size_t)(m0 +      mr + r) * NEXP + n0      + ml] = acc00[r];
        logits[(size_t)(m0 +      mr + r) * NEXP + n0 + 16 + ml] = acc01[r];
        logits[(size_t)(m0 + 16 + mr + r) * NEXP + n0      + ml] = acc10[r];
        logits[(size_t)(m0 + 16 + mr + r) * NEXP + n0 + 16 + ml] = acc11[r];
    }
}

// ---------------------------------------------------------------------------
// Kernel 2: per-token routing. One wave32 per token; 8 experts per lane.
// ---------------------------------------------------------------------------
__global__ __launch_bounds__(256) void router_topk(
    const float* __restrict__ logits, const float* __restrict__ bias,
    float* __restrict__ topw, int* __restrict__ topi, int T)
{
    const int lane = threadIdx.x & 31;
    const int wv   = threadIdx.x >> 5;
    const int t    = blockIdx.x * 8 + wv;
    if (t >= T) return;

    const float* row = logits + (size_t)t * NEXP;

    float sc[8], ch[8];
    #pragma unroll
    for (int j = 0; j < 8; ++j) {
        const int e = lane + 32 * j;
        const float s = 1.0f / (1.0f + __expf(-row[e]));
        sc[j] = s;
        ch[j] = s + bias[e];
    }

    float myW = 0.0f, sum = 0.0f;
    int   myI = 0;

    #pragma unroll
    for (int i = 0; i < TOPK; ++i) {
        float bv = ch[0]; int bj = 0;
        #pragma unroll
        for (int j = 1; j < 8; ++j)
            if (ch[j] > bv) { bv = ch[j]; bj = j; }
        int bi = lane + 32 * bj;

        #pragma unroll
        for (int off = 16; off >= 1; off >>= 1) {
            const float ov = __shfl_xor(bv, off, 32);
            const int   oi = __shfl_xor(bi, off, 32);
            if (ov > bv || (ov == bv && oi < bi)) { bv = ov; bi = oi; }
        }
        const int jj = bi >> 5;              // uniform
        const int ol = bi & 31;              // uniform
        float sl = sc[0];
        #pragma unroll
        for (int j = 1; j < 8; ++j)
            if (jj == j) sl = sc[j];
        const float raw = __shfl(sl, ol, 32);

        #pragma unroll
        for (int j = 0; j < 8; ++j)
            if (lane == ol && j == jj) ch[j] = -INFINITY;

        sum += raw;
        if (lane == i) { myW = raw; myI = bi; }
    }

    if (lane < TOPK) {
        topw[(size_t)t * TOPK + lane] = myW / (sum + 1e-20f);
        topi[(size_t)t * TOPK + lane] = myI;
    }
}

// ---------------------------------------------------------------------------
extern "C" void kernel_launch(void* const* d_in, const int* in_sizes, int n_in,
                              void* d_out, int out_size, void* d_ws, size_t ws_size,
                              hipStream_t stream)
{
    const float* H    = (const float*)d_in[0];   // [T, 4096]
    const float* W    = (const float*)d_in[1];   // [256, 4096]
    const float* bias = (const float*)d_in[2];   // [256]

    const int T = in_sizes[0] / HIDDEN;          // 8192

    float* logits = (float*)d_out;
    float* topw   = logits + (size_t)T * NEXP;
    int*   topi   = (int*)(topw + (size_t)T * TOPK);

    gate_gemm_f32<<<T / BM, 256, 0, stream>>>(H, W, logits, T);
    router_topk<<<(T + 7) / 8, 256, 0, stream>>>(logits, bias, topw, topi, T);
}